// PagedAttention_27084063769100
// MI455X (gfx1250) — compile-verified
//
#include <hip/hip_runtime.h>
#include <math.h>

// PagedAttention decode, fp32, for gfx1250 (MI455X).
// 64 seqs x 8 kv-heads; 4 queries per kv head (GQA); head 128; block 16.

#define NKVH  8
#define GQ    4
#define HD    128
#define BS    16
#define MBS   128
#define SCALE 0.08838834764831845f
#define LOG2E 1.4426950408889634f

typedef float v2f __attribute__((ext_vector_type(2)));
typedef float v4f __attribute__((ext_vector_type(4)));
typedef float v8f __attribute__((ext_vector_type(8)));
typedef int   v4i __attribute__((ext_vector_type(4)));

#define GAS __attribute__((address_space(1)))  // global ("__device__")
#define LAS __attribute__((address_space(3)))  // LDS

#if defined(__has_builtin)
#if __has_builtin(__builtin_amdgcn_global_load_async_to_lds_b128)
#define USE_ASYNC 1
#endif
#endif
#ifndef USE_ASYNC
#define USE_ASYNC 0
#endif

static __device__ __forceinline__ v8f wmma_f32(v2f a, v2f b, v8f c) {
  // D(16x16) = A(16x4) * B(4x16) + C   -- v_wmma_f32_16x16x4_f32
  return __builtin_amdgcn_wmma_f32_16x16x4_f32(false, a, false, b, (short)0, c,
                                               false, false);
}

__global__ __launch_bounds__(128) void paged_attn_wmma(
    const float* __restrict__ q,
    const float* __restrict__ knew_all,
    const float* __restrict__ vnew_all,
    const float* __restrict__ kcache,
    const float* __restrict__ vcache,
    const int* __restrict__ btab,
    const int* __restrict__ ctxlen,
    float* __restrict__ out) {
  const int h    = blockIdx.x;       // kv head
  const int s    = blockIdx.y;       // sequence
  const int tid  = threadIdx.x;
  const int w    = tid >> 5;         // wave 0..3
  const int lane = tid & 31;
  const int half = lane >> 4;        // 0 = lanes 0-15, 1 = lanes 16-31
  const int lm   = lane & 15;

  // LDS layout (floats): Qs[16x132] | Vw[4][16x132] | ML[4][8] | AC[4][4][128]
  __shared__ float lds[12640];
  float* Qs = lds;                    // padded stride 132 -> bank-conflict free
  float* Vw = lds + 2112 + w * 2112;  // private per wave
  float* ML = lds + 2112 * 5;
  float* AC = ML + 32;

  const int ctx    = ctxlen[s];
  const int ntiles = (ctx + BS - 1) >> 4;
  const int newpos = ctx - 1;         // the new token's position

  // Stage Q, pre-scaled by SCALE*log2(e); rows 4..15 zero so padded query
  // columns of the WMMA stay finite.
  for (int i = tid; i < 16 * 132; i += 128) {
    int r = i / 132;
    int c = i - r * 132;
    float v = 0.0f;
    if (r < GQ && c < HD)
      v = q[((size_t)s * 32 + h * GQ + r) * HD + c] * (SCALE * LOG2E);
    Qs[i] = v;
  }
  __syncthreads();

  const float* knew = knew_all + ((size_t)s * NKVH + h) * HD;
  const float* vnew = vnew_all + ((size_t)s * NKVH + h) * HD;
  const v4f vnew4 = ((const v4f*)vnew)[lane];  // lane*4 .. lane*4+3

  float m_run = -__builtin_inff();
  float l_run = 0.0f;
  v8f acc[8];  // out^T accumulator: dim = c*16 + r + 8*half, query = lane%16
#pragma unroll
  for (int c = 0; c < 8; ++c) acc[c] = (v8f){0, 0, 0, 0, 0, 0, 0, 0};

  for (int tile = w; tile < ntiles; tile += 4) {
    const int blk = btab[s * MBS + tile];                // one block == one tile
    const size_t boff = ((size_t)blk * NKVH + h) << 11;  // * 16 * 128
    const float* kblk = kcache + boff;
    const float* vblk = vcache + boff;
    const int tok0 = tile << 4;

    // ---- stage V tile (16x128) into LDS, row stride 132 ----
#if USE_ASYNC
    {
      const float* gsrc = vblk + lane * 4;
      float* ldst = Vw + lane * 4;
#pragma unroll
      for (int j = 0; j < 16; ++j) {
        __builtin_amdgcn_global_load_async_to_lds_b128(
            (GAS v4i*)(gsrc + j * HD), (LAS v4i*)(ldst + j * 132), 0, 0);
      }
#if __has_builtin(__builtin_amdgcn_s_wait_asynccnt)
      __builtin_amdgcn_s_wait_asynccnt(0);
#else
      asm volatile("s_wait_asynccnt 0" ::: "memory");
#endif
      asm volatile("" ::: "memory");
    }
#else
    {
#pragma unroll
      for (int j = 0; j < 16; ++j) {
        v4f t = __builtin_nontemporal_load(
            (const v4f*)(vblk + j * HD + lane * 4));
        *(v4f*)(Vw + j * 132 + lane * 4) = t;
      }
    }
#endif
    // Override the new token's V row (wave-uniform branch; same-wave DS is
    // in order, so no barrier needed).
    if (newpos >= tok0 && newpos < tok0 + BS) {
      *(v4f*)(Vw + (newpos & 15) * 132 + lane * 4) = vnew4;
    }

    // ---- per-lane K row pointer in A-operand layout, new-token override ----
    const float* krow = kblk + lm * HD + half * 2;
    if (tok0 + lm == newpos) krow = knew + half * 2;
    const float* qrow = Qs + lm * 132 + half * 2;

    // ---- S^T[token, query] = K_tile x Q^T : 32 x wmma f32 16x16x4 ----
    v8f st = (v8f){0, 0, 0, 0, 0, 0, 0, 0};
#pragma unroll
    for (int kk = 0; kk < 32; ++kk) {
      v2f a = __builtin_nontemporal_load((const v2f*)(krow + kk * 4));
      v2f b = *(const v2f*)(qrow + kk * 4);
      st = wmma_f32(a, b, st);
    }

    // ---- online softmax in base-2 domain ----
    float sv[8];
#pragma unroll
    for (int r = 0; r < 8; ++r) {
      int tokid = tok0 + r + half * 8;  // token = VGPR r + 8*half
      sv[r] = (tokid < ctx) ? st[r] : -__builtin_inff();
    }
    float mt = sv[0];
#pragma unroll
    for (int r = 1; r < 8; ++r) mt = fmaxf(mt, sv[r]);
    mt = fmaxf(mt, __shfl_xor(mt, 16, 32));  // combine lane halves
    const float m_new = fmaxf(m_run, mt);
    const float scl = exp2f(m_run - m_new);
    m_run = m_new;
    float p[8];
    float lt = 0.0f;
#pragma unroll
    for (int r = 0; r < 8; ++r) {
      p[r] = exp2f(sv[r] - m_new);
      lt += p[r];
    }
    lt += __shfl_xor(lt, 16, 32);
    l_run = l_run * scl + lt;
#pragma unroll
    for (int c = 0; c < 8; ++c)
#pragma unroll
      for (int r = 0; r < 8; ++r) acc[c][r] *= scl;

    // ---- acc^T += V^T x P^T : 4 token-steps x 8 dim-chunks of wmma ----
#pragma unroll
    for (int kk = 0; kk < 4; ++kk) {
      const int r0 = (kk & 1) << 2;
      // Build B(4x16) = P^T rows 4kk..4kk+3 from S^T registers: values for
      // the other lane-half come via shfl_xor(16).
      const float sw0 = __shfl_xor(p[r0 + ((kk < 2) ? 2 : 0)], 16, 32);
      const float sw1 = __shfl_xor(p[r0 + ((kk < 2) ? 3 : 1)], 16, 32);
      v2f b;
      if (kk < 2) {
        b.x = half ? sw0 : p[r0 + 0];
        b.y = half ? sw1 : p[r0 + 1];
      } else {
        b.x = half ? p[r0 + 2] : sw0;
        b.y = half ? p[r0 + 3] : sw1;
      }
      const float* vptr = Vw + (kk * 4 + half * 2) * 132 + lm;
#pragma unroll
      for (int c = 0; c < 8; ++c) {
        v2f a;  // A = V^T: lane lm = dim within chunk c; k = token
        a.x = vptr[c * 16];
        a.y = vptr[c * 16 + 132];
        acc[c] = wmma_f32(a, b, acc[c]);
      }
    }
  }

  // ---- publish per-wave partials ----
  if (lane < GQ) {
    ML[w * 8 + lane] = m_run;       // per-query running max (base-2)
    ML[w * 8 + 4 + lane] = l_run;   // per-query running sum
  }
  if (lm < GQ) {
#pragma unroll
    for (int c = 0; c < 8; ++c)
#pragma unroll
      for (int r = 0; r < 8; ++r)
        AC[(w * GQ + lm) * HD + c * 16 + r + half * 8] = acc[c][r];
  }
  __syncthreads();

  // ---- cross-wave combine; thread tid owns dim d = tid ----
#pragma unroll
  for (int qi = 0; qi < GQ; ++qi) {
    float M = ML[qi];
#pragma unroll
    for (int w2 = 1; w2 < 4; ++w2) M = fmaxf(M, ML[w2 * 8 + qi]);
    float lsum = 0.0f, o = 0.0f;
#pragma unroll
    for (int w2 = 0; w2 < 4; ++w2) {
      const float e = exp2f(ML[w2 * 8 + qi] - M);
      lsum += ML[w2 * 8 + 4 + qi] * e;
      o += AC[(w2 * GQ + qi) * HD + tid] * e;
    }
    out[((size_t)s * 32 + h * GQ + qi) * HD + tid] = o / lsum;
  }
}

extern "C" void kernel_launch(void* const* d_in, const int* in_sizes, int n_in,
                              void* d_out, int out_size, void* d_ws,
                              size_t ws_size, hipStream_t stream) {
  (void)in_sizes; (void)n_in; (void)out_size; (void)d_ws; (void)ws_size;
  const float* q  = (const float*)d_in[0];
  const float* kn = (const float*)d_in[1];
  const float* vn = (const float*)d_in[2];
  const float* kc = (const float*)d_in[3];
  const float* vc = (const float*)d_in[4];
  const int* bt   = (const int*)d_in[5];
  const int* cl   = (const int*)d_in[6];
  // d_in[7] (slot_mapping) is implied by context_lens/block_tables; the new
  // token is folded in in-register so the caches are never mutated.
  dim3 grid(NKVH, 64);
  dim3 block(128);
  paged_attn_wmma<<<grid, block, 0, stream>>>(q, kn, vn, kc, vc, bt, cl,
                                              (float*)d_out);
}